// MyModel_87522843559687
// MI455X (gfx1250) — compile-verified
//
#include <hip/hip_runtime.h>
#include <hip/hip_bf16.h>

typedef __attribute__((ext_vector_type(2))) float v2f;
typedef __attribute__((ext_vector_type(8))) float v8f;

#define BATCH 256
#define TSTEPS 2048
#define FEAT 64
#define UNITS 10
#define G4 40           // 4*UNITS
#define NTILES 4        // 4 gate tiles of 16 cols (padded)
#define WTILES 16       // 256 / 16 batch-row tiles

// xz scratch layout: [w][t][gate][lane][8]  (float), 32B contiguous per lane
__device__ __forceinline__ size_t xz_index(int w, int t, int g, int lane) {
    return ((((size_t)w * TSTEPS + t) * NTILES + g) * 32 + lane) * 8;
}

// ---------------------------------------------------------------------------
// Phase 1: xz[w][t][gate-tile] = x[16w..16w+15, t, :] @ kernel_ext + bias_ext
// A: 16x4 f32 per k-step, B: 4x16 f32 (register-resident), C/D: 16x16 f32.
// launch_bounds(...,1): allow full VGPR budget so B (128 VGPRs) never spills.
// ---------------------------------------------------------------------------
__global__ void __launch_bounds__(256, 1)
lstm_pre_kernel(const float* __restrict__ x,
                const float* __restrict__ kernel,
                const float* __restrict__ bias,
                float* __restrict__ xz) {
    // Packed kernel_ext in LDS: pair (k, k+1) interleaved so each B fragment
    // (2 VGPRs) is a single ds_load_b64:  kl[(k>>1)*128 + e*2 + (k&1)]
    __shared__ float kl[32 * 128];   // 16 KB

    const int tid = threadIdx.x;
    for (int idx = tid; idx < FEAT * 64; idx += blockDim.x) {
        int k = idx >> 6;          // 0..63
        int e = idx & 63;          // ext col
        int g = e >> 4, j = e & 15;
        float v = (j < UNITS) ? kernel[k * G4 + g * UNITS + j] : 0.0f;
        kl[(k >> 1) * 128 + e * 2 + (k & 1)] = v;
    }
    __syncthreads();

    const int w      = blockIdx.x;         // batch tile
    const int lane   = tid & 31;
    const int waveid = tid >> 5;           // 0..7
    const int hi     = lane >> 4;          // 0 / 1
    const int lm     = lane & 15;

    // B fragments register-resident for the whole kernel: [gate][ks]
    v2f bfr[NTILES][16];
#pragma unroll
    for (int g = 0; g < NTILES; ++g) {
        const int e = g * 16 + lm;
#pragma unroll
        for (int ks = 0; ks < 16; ++ks)
            bfr[g][ks] = *(const v2f*)(&kl[(2 * ks + hi) * 128 + e * 2]);
    }

    // per-column bias (same for every row of a tile)
    float cbias[NTILES];
#pragma unroll
    for (int g = 0; g < NTILES; ++g)
        cbias[g] = (lm < UNITS) ? bias[g * UNITS + lm] : 0.0f;

    const int t0 = blockIdx.y * 64 + waveid;   // 8 waves, stride 8, 8 iters

#pragma unroll 1
    for (int ti = 0; ti < 8; ++ti) {
        const int t = t0 + ti * 8;

        // A fragments for all 16 k-steps: lane -> row (lm), K = 4*ks + 2*hi (+1)
        v2f a[16];
        const float* xrow = x + (((size_t)(16 * w + lm)) * TSTEPS + t) * FEAT + 2 * hi;
#pragma unroll
        for (int ks = 0; ks < 16; ++ks)
            a[ks] = *(const v2f*)(xrow + 4 * ks);

#pragma unroll
        for (int g = 0; g < NTILES; ++g) {
            v8f c;
#pragma unroll
            for (int r = 0; r < 8; ++r) c[r] = cbias[g];
#pragma unroll
            for (int ks = 0; ks < 16; ++ks)
                c = __builtin_amdgcn_wmma_f32_16x16x4_f32(
                    false, a[ks], false, bfr[g][ks], (short)0, c, false, false);
            *(v8f*)(xz + xz_index(w, t, g, lane)) = c;
        }
    }
}

// ---------------------------------------------------------------------------
// Phase 2: sequential LSTM scan. 16 workgroups x 1 wave; each wave owns 16
// batch rows. h,c live in registers in WMMA C-layout; gate tiles align so the
// cell update is pure per-lane VALU. h -> A-layout via a 1 KB LDS bounce.
// ---------------------------------------------------------------------------
__global__ void __launch_bounds__(32, 1)
lstm_scan_kernel(const float* __restrict__ xz,
                 const float* __restrict__ rker,      // [10,40]
                 const float* __restrict__ dense_w,   // [10]
                 const float* __restrict__ dense_b,   // [1]
                 float* __restrict__ out) {
    __shared__ float hbuf[16 * 16];

    const int w    = blockIdx.x;
    const int lane = threadIdx.x & 31;
    const int hi   = lane >> 4;
    const int lm   = lane & 15;

    // Resident B fragments of recurrent_kernel_ext: [gate][ks] (K padded 10->12)
    v2f rb[NTILES][3];
#pragma unroll
    for (int g = 0; g < NTILES; ++g)
#pragma unroll
        for (int ks = 0; ks < 3; ++ks) {
            int k0 = 4 * ks + 2 * hi;
            float b0 = (k0 < UNITS && lm < UNITS) ? rker[k0 * G4 + g * UNITS + lm] : 0.0f;
            float b1 = (k0 + 1 < UNITS && lm < UNITS) ? rker[(k0 + 1) * G4 + g * UNITS + lm] : 0.0f;
            rb[g][ks] = (v2f){b0, b1};
        }

    v8f hreg, creg;
#pragma unroll
    for (int r = 0; r < 8; ++r) { hreg[r] = 0.0f; creg[r] = 0.0f; }

#pragma unroll 1
    for (int t = 0; t < TSTEPS; ++t) {
        // C-layout h -> LDS at [m][u]
#pragma unroll
        for (int r = 0; r < 8; ++r)
            hbuf[(r + 8 * hi) * 16 + lm] = hreg[r];
        __syncthreads();   // single-wave WG: S_NOP + waitcnt

        // A fragments: lane -> row lm, K = 4*ks + 2*hi (+1)
        v2f a[3];
#pragma unroll
        for (int ks = 0; ks < 3; ++ks)
            a[ks] = *(const v2f*)(&hbuf[lm * 16 + 4 * ks + 2 * hi]);
        __syncthreads();

        // z_t from precomputed projection (accumulator init)
        v8f z[NTILES];
#pragma unroll
        for (int g = 0; g < NTILES; ++g)
            z[g] = *(const v8f*)(xz + xz_index(w, t, g, lane));

#pragma unroll
        for (int g = 0; g < NTILES; ++g)
#pragma unroll
            for (int ks = 0; ks < 3; ++ks)
                z[g] = __builtin_amdgcn_wmma_f32_16x16x4_f32(
                    false, a[ks], false, rb[g][ks], (short)0, z[g], false, false);

        // gates: i,f,o sigmoid; g softsign; same-lane combine across tiles
#pragma unroll
        for (int r = 0; r < 8; ++r) {
            float ig = 1.0f / (1.0f + __expf(-z[0][r]));
            float fg = 1.0f / (1.0f + __expf(-z[1][r]));
            float gg = z[2][r] / (1.0f + fabsf(z[2][r]));
            float og = 1.0f / (1.0f + __expf(-z[3][r]));
            float cv = fg * creg[r] + ig * gg;
            creg[r] = cv;
            hreg[r] = og * (cv / (1.0f + fabsf(cv)));
        }
    }

    // epilogue: out[b] = sigmoid(h[b] . dense_w + dense_b)
#pragma unroll
    for (int r = 0; r < 8; ++r)
        hbuf[(r + 8 * hi) * 16 + lm] = hreg[r];
    __syncthreads();

    if (lane < 16) {
        float acc = dense_b[0];
#pragma unroll
        for (int u = 0; u < UNITS; ++u)
            acc += hbuf[lane * 16 + u] * dense_w[u];
        out[w * 16 + lane] = 1.0f / (1.0f + __expf(-acc));
    }
}

extern "C" void kernel_launch(void* const* d_in, const int* in_sizes, int n_in,
                              void* d_out, int out_size, void* d_ws, size_t ws_size,
                              hipStream_t stream) {
    (void)in_sizes; (void)n_in; (void)out_size; (void)ws_size;
    const float* x       = (const float*)d_in[0];
    const float* kernel  = (const float*)d_in[1];
    const float* rkernel = (const float*)d_in[2];
    const float* bias    = (const float*)d_in[3];
    const float* dense_w = (const float*)d_in[4];
    const float* dense_b = (const float*)d_in[5];
    float* out = (float*)d_out;
    float* xz  = (float*)d_ws;   // 16*2048*4*32*8 floats = 128 MiB

    lstm_pre_kernel<<<dim3(WTILES, 32), 256, 0, stream>>>(x, kernel, bias, xz);
    lstm_scan_kernel<<<WTILES, 32, 0, stream>>>(xz, rkernel, dense_w, dense_b, out);
}